// DNPP_82497731822005
// MI455X (gfx1250) — compile-verified
//
#include <hip/hip_runtime.h>

// ---------------------------------------------------------------------------
// Fused: segment_sum(E->N) ∘ segment_sum(N->B) ∘ mean ∘ linear  ==
//   out[g,o] = (1/cnt_g) * sum_{e: batch[edge_idx[e]]==g} <emb[e], W[o]> + b[o]
// Single streaming pass over edge_embedding (1.536 GB -> ~66us @ 23.3 TB/s).
// Per-edge dots computed on the matrix pipe with V_WMMA_F32_16X16X4_F32:
//   A = 16-edge x 192 f32 tile, B = W^T zero-padded to 192x16.
// Streaming inputs use non-temporal loads so batch/W stay resident in cache.
// ---------------------------------------------------------------------------

typedef __attribute__((ext_vector_type(2))) float v2f;
typedef __attribute__((ext_vector_type(8))) float v8f;

#define DCH    192
#define NOUT   3
#define NGRAPH 64
#define TILE   16
#define PADK   196   // 196 % 64 == 4 -> LDS b64 reads of W hit all 64 banks once

// ---- kernel 0: zero workspace (gAcc[64*3] + counts[64]) --------------------
__global__ void k_zero(float* __restrict__ ws) {
    int i = blockIdx.x * blockDim.x + threadIdx.x;
    if (i < NGRAPH * NOUT + NGRAPH) ws[i] = 0.0f;
}

// ---- kernel 1: nodes-per-graph histogram ----------------------------------
__global__ void k_count(const int* __restrict__ batch, int n,
                        float* __restrict__ counts) {
    __shared__ float h[NGRAPH];
    int tid = threadIdx.x;
    if (tid < NGRAPH) h[tid] = 0.0f;
    __syncthreads();
    for (int i = blockIdx.x * blockDim.x + tid; i < n;
         i += gridDim.x * blockDim.x) {
        unsafeAtomicAdd(&h[batch[i]], 1.0f);
    }
    __syncthreads();
    if (tid < NGRAPH && h[tid] != 0.0f) unsafeAtomicAdd(&counts[tid], h[tid]);
}

// ---- kernel 2: main streaming WMMA pass over edges ------------------------
__global__ void __launch_bounds__(256)
k_edges(const float* __restrict__ emb,   // [E, 192]
        const int*   __restrict__ eidx,  // [E]
        const int*   __restrict__ batch, // [N]
        const float* __restrict__ W,     // [3, 192]
        float*       __restrict__ gAcc,  // [64*3] global accumulator
        int E) {
    __shared__ float sW[TILE][PADK];     // W^T padded to 16 rows (rows 3..15 = 0)
    __shared__ float sAcc[NGRAPH][4];    // per-block [graph][out] accumulator

    const int tid = threadIdx.x;

    // Stage padded W into LDS (12.5 KB, read 48x per tile per wave).
    for (int i = tid; i < TILE * DCH; i += blockDim.x) {
        int n = i / DCH, k = i % DCH;
        sW[n][k] = (n < NOUT) ? W[n * DCH + k] : 0.0f;
    }
    for (int i = tid; i < NGRAPH * 4; i += blockDim.x)
        (&sAcc[0][0])[i] = 0.0f;
    __syncthreads();

    const int lane  = tid & 31;
    const int wave  = tid >> 5;
    const int hi    = lane >> 4;          // K sub-offset selector (ISA A/B layout)
    const int nn    = lane & 15;          // A: row (M); B/D: column (N)
    const bool act  = (nn < NOUT);        // lanes owning useful D columns
    const int gwave = blockIdx.x * (blockDim.x >> 5) + wave;
    const int nwave = gridDim.x * (blockDim.x >> 5);
    const int numTiles = (E + TILE - 1) / TILE;

    for (int t = gwave; t < numTiles; t += nwave) {
        const long base = (long)t * TILE;

        // graph id for edge m (held by lane m, m = 0..15); stream eidx as NT.
        int g = -1;
        if (lane < TILE) {
            long e = base + lane;
            if (e < E) g = batch[__builtin_nontemporal_load(eidx + e)];
        }

        // Hoist the wave broadcasts of g out of the K loop: gm[v] is the
        // graph id of D-matrix row M = v + 8*hi owned by this lane.
        int gm[8];
        #pragma unroll
        for (int v = 0; v < 8; ++v) gm[v] = __shfl(g, v + 8 * hi, 32);

        // A-fragment row pointer: lane reads emb[base+nn][k + 2*hi .. +1]
        long row = base + nn; if (row >= E) row = E - 1;   // tail guard
        const float* rowp = emb + row * (long)DCH + 2 * hi;

        v8f c = {0.f, 0.f, 0.f, 0.f, 0.f, 0.f, 0.f, 0.f};
        #pragma unroll 8
        for (int k = 0; k < DCH; k += 4) {
            v2f a = __builtin_nontemporal_load((const v2f*)(rowp + k)); // NT stream
            v2f b = *(const v2f*)(&sW[nn][k + 2 * hi]);      // ds_load_b64, bank-free
            c = __builtin_amdgcn_wmma_f32_16x16x4_f32(
                    false, a, false, b, (short)0, c, false, false);
        }

        // D layout: VGPR v, lane l -> M = v + 8*(l/16), N = l%16.
        // Columns 0..2 hold <emb[edge], W[n]>. 3 LDS atomics per edge.
        #pragma unroll
        for (int v = 0; v < 8; ++v) {
            if (act && gm[v] >= 0)
                unsafeAtomicAdd(&sAcc[gm[v]][nn], c[v]);
        }
    }

    __syncthreads();
    // Flush block accumulator: 192 global float atomics per block.
    for (int i = tid; i < NGRAPH * NOUT; i += blockDim.x) {
        int gg = i / NOUT, o = i % NOUT;
        float v = sAcc[gg][o];
        if (v != 0.0f) unsafeAtomicAdd(&gAcc[i], v);
    }
}

// ---- kernel 3: finalize (mean + bias) -------------------------------------
__global__ void k_final(const float* __restrict__ ws,
                        const float* __restrict__ bias,
                        float* __restrict__ out) {
    int i = threadIdx.x;
    if (i < NGRAPH * NOUT) {
        int g = i / NOUT, o = i % NOUT;
        float cnt = ws[NGRAPH * NOUT + g];
        out[i] = ws[i] / fmaxf(cnt, 1.0f) + bias[o];
    }
}

// ---------------------------------------------------------------------------
extern "C" void kernel_launch(void* const* d_in, const int* in_sizes, int n_in,
                              void* d_out, int out_size, void* d_ws, size_t ws_size,
                              hipStream_t stream) {
    const float* emb   = (const float*)d_in[0];  // [E, 192]
    const int*   eidx  = (const int*)  d_in[1];  // [E]
    const int*   batch = (const int*)  d_in[2];  // [N]
    const float* W     = (const float*)d_in[3];  // [3, 192]
    const float* bias  = (const float*)d_in[4];  // [3]
    const int E = in_sizes[1];
    const int N = in_sizes[2];

    float* ws     = (float*)d_ws;                 // [0..191]=gAcc, [192..255]=counts
    float* gAcc   = ws;
    float* counts = ws + NGRAPH * NOUT;

    k_zero<<<1, 256, 0, stream>>>(ws);
    k_count<<<128, 256, 0, stream>>>(batch, N, counts);
    k_edges<<<1024, 256, 0, stream>>>(emb, eidx, batch, W, gAcc, E);
    k_final<<<1, 256, 0, stream>>>(ws, bias, (float*)d_out);
}